// MyMaxPool_7490422964872
// MI455X (gfx1250) — compile-verified
//
#include <hip/hip_runtime.h>
#include <cstdint>

// MaxNetwork 2x2/stride-2 pooling, C=64 H=512 W=512 -> [64,256,256], fp32.
// Memory-bound: 84 MB total traffic => ~3.6us at 23.3 TB/s. Strategy:
//   - One block per (channel, 4 output rows): 4096 blocks x 256 threads.
//   - TDM (tensor_load_to_lds) DMAs the 8x512 f32 input tile (16 KB) into LDS.
//   - s_wait_tensorcnt 0 + workgroup barrier, then each thread computes 4
//     outputs from LDS and writes one float4 with a non-temporal store.

typedef __attribute__((ext_vector_type(4))) unsigned int v4u;
typedef __attribute__((ext_vector_type(8))) int          v8i;
typedef __attribute__((ext_vector_type(4))) int          v4i;
typedef __attribute__((ext_vector_type(4))) float        v4f;

#define NC 64
#define NH 512
#define NW 512
#define NOH 256
#define NOW 256
#define ROWS_PER_BLK 4          // output rows per block
#define IN_ROWS (2 * ROWS_PER_BLK)
#define TILE_ELEMS (IN_ROWS * NW)   // 4096 floats = 16 KB LDS

__device__ __forceinline__ float pairmax(float a, float b) {
    // relu(a-b) + relu(b); outer relu is identity (sum of two relus >= 0).
    return fmaxf(a - b, 0.0f) + fmaxf(b, 0.0f);
}

__global__ __launch_bounds__(256) void maxnet_pool_tdm(const float* __restrict__ x,
                                                       float* __restrict__ out) {
    __shared__ float tile[TILE_ELEMS];

    const unsigned bid    = blockIdx.x;      // 0..4095
    const unsigned c      = bid >> 6;        // channel
    const unsigned rg     = bid & 63;        // row group within channel
    const unsigned ohbase = rg * ROWS_PER_BLK;
    const unsigned ihbase = ohbase * 2;

    // ---- TDM: DMA the 8x512 f32 tile from global into LDS (wave 0 only) ----
    if (threadIdx.x < 32) {
        const unsigned long long gaddr =
            (unsigned long long)(uintptr_t)(x + ((size_t)c * NH + ihbase) * NW);
        const unsigned lds_off = (unsigned)(uintptr_t)(&tile[0]); // LDS byte offset

        v4u g0;
        g0.x = 1u;                                        // count=1 (valid user D#)
        g0.y = lds_off;                                   // lds_addr [63:32]
        g0.z = (unsigned)(gaddr & 0xFFFFFFFFull);         // global_addr[31:0]
        g0.w = (unsigned)((gaddr >> 32) & 0x1FFFFFFull)   // global_addr[56:32]
             | (2u << 30);                                // type = 2 ("image")

        v8i g1;
        g1[0] = (int)(2u << 16);        // workgroup_mask=0, data_size=2 (4 bytes)
        g1[1] = (int)(512u << 16);      // tensor_dim0[15:0]=512 in bits[31:16]
        g1[2] = (int)(8u << 16);        // tensor_dim0 hi=0; tensor_dim1[15:0]=8
        g1[3] = (int)(512u << 16);      // tensor_dim1 hi=0; tile_dim0=512
        g1[4] = 8;                      // tile_dim1=8; tile_dim2=0
        g1[5] = 512;                    // tensor_dim0_stride[31:0]=512 elements
        g1[6] = 0;                      // stride0 hi; tensor_dim1_stride lo
        g1[7] = 0;                      // tensor_dim1_stride hi

        v4i g2 = {0, 0, 0, 0};          // 2D tensor: groups 2/3 unused
        v4i g3 = {0, 0, 0, 0};
        v8i g4 = {0, 0, 0, 0, 0, 0, 0, 0};

        __builtin_amdgcn_tensor_load_to_lds(g0, g1, g2, g3, g4, 0);
        __builtin_amdgcn_s_wait_tensorcnt(0);
    }
    __syncthreads();

    // ---- Compute: thread t -> output row r (0..3), col group g (4 outputs) ----
    const unsigned t = threadIdx.x;
    const unsigned r = t >> 6;          // 0..3
    const unsigned g = t & 63;          // 0..63 -> output cols 4g..4g+3

    const v4f a0 = *(const v4f*)&tile[(2 * r)     * NW + 8 * g];
    const v4f a1 = *(const v4f*)&tile[(2 * r)     * NW + 8 * g + 4];
    const v4f b0 = *(const v4f*)&tile[(2 * r + 1) * NW + 8 * g];
    const v4f b1 = *(const v4f*)&tile[(2 * r + 1) * NW + 8 * g + 4];

    v4f res;
    res.x = pairmax(pairmax(a0.x, a0.y), pairmax(b0.x, b0.y));
    res.y = pairmax(pairmax(a0.z, a0.w), pairmax(b0.z, b0.w));
    res.z = pairmax(pairmax(a1.x, a1.y), pairmax(b1.x, b1.y));
    res.w = pairmax(pairmax(a1.z, a1.w), pairmax(b1.z, b1.w));

    v4f* dst = (v4f*)(out + ((size_t)c * NOH + (ohbase + r)) * NOW + 4 * g);
    __builtin_nontemporal_store(res, dst);   // streamed output, bypass near caches
}

extern "C" void kernel_launch(void* const* d_in, const int* in_sizes, int n_in,
                              void* d_out, int out_size, void* d_ws, size_t ws_size,
                              hipStream_t stream) {
    (void)in_sizes; (void)n_in; (void)d_ws; (void)ws_size; (void)out_size;
    const float* x = (const float*)d_in[0];
    float* out = (float*)d_out;
    dim3 grid(NC * (NOH / ROWS_PER_BLK));   // 64 * 64 = 4096
    dim3 block(256);
    maxnet_pool_tdm<<<grid, block, 0, stream>>>(x, out);
}